// SimpleDescriptor_30374008717894
// MI455X (gfx1250) — compile-verified
//
#include <hip/hip_runtime.h>
#include <cstdint>
#include <cstddef>

// ---------------- problem constants ----------------
#define PATCH       224
#define IMG_W       4096
#define NTHREADS    256              // 8 waves of 32 on gfx1250
#define ROWS        16               // patch rows per pipeline stage
#define ROW_PAD     228              // floats per LDS row (16B-aligned superset of 224)
#define STAGES      (PATCH / ROWS)   // 14
#define TPS         (ROWS * (ROW_PAD / 4))   // 912 b128 transfers per stage
#define ISSUE_ITERS (TPS / NTHREADS + 1)     // 4 async instrs per wave per stage
#define PATCH_ELEMS (PATCH * PATCH)          // 50176

// ---------------- CDNA5 async global->LDS path ----------------
#if defined(__has_builtin)
#  if __has_builtin(__builtin_amdgcn_global_load_async_to_lds_b128)
#    define USE_ASYNC_BUILTIN 1
#  endif
#  if __has_builtin(__builtin_amdgcn_s_wait_asynccnt)
#    define USE_WAIT_BUILTIN 1
#  endif
#endif

#if defined(USE_WAIT_BUILTIN)
#  define WAIT_ASYNC(n) __builtin_amdgcn_s_wait_asynccnt(n)
#else
#  define WAIT_ASYNC(n) asm volatile("s_wait_asynccnt %0" ::"i"(n) : "memory")
#endif

// Builtin signature (per hipcc diagnostic): param 1 = v4i in global AS,
// param 2 = v4i in LDS AS, then two immediate ints (offset, cpol).
typedef int v4i __attribute__((ext_vector_type(4)));
typedef v4i __attribute__((address_space(1))) *gv4i_p;
typedef v4i __attribute__((address_space(3))) *lv4i_p;

// Copy 16 bytes from global to LDS via the async copy engine (ASYNCcnt).
__device__ __forceinline__ void async_copy_b128(const float* src, float* dst_lds) {
#if defined(USE_ASYNC_BUILTIN)
  __builtin_amdgcn_global_load_async_to_lds_b128(
      (gv4i_p)(uintptr_t)src, (lv4i_p)(uint32_t)(uintptr_t)dst_lds,
      /*offset=*/0, /*cpol=*/0);
#else
  // VDST = 32-bit LDS byte address, VADDR = 64-bit global address, saddr=off
  uint32_t laddr = (uint32_t)(uintptr_t)dst_lds;
  asm volatile("global_load_async_to_lds_b128 %0, %1, off"
               ::"v"(laddr), "v"(src)
               : "memory");
#endif
}

// ---------------- kernel ----------------
__global__ __launch_bounds__(NTHREADS) void patch_stats_kernel(
    const float* __restrict__ image, const int* __restrict__ coords_x,
    const int* __restrict__ coords_y, float* __restrict__ out) {
  const int n   = blockIdx.x;
  const int tid = threadIdx.x;

  const int x_low = coords_x[n] - PATCH / 2;   // in [0, 3871]
  const int y_low = coords_y[n] - PATCH / 2;   // in [0, 3871]
  const int xa    = x_low & ~3;                // 16B-aligned start column
  const int d     = x_low & 3;                 // leading floats to skip in LDS

  __shared__ float lds[2][ROWS * ROW_PAD];     // 2 x 14592 B double buffer
  __shared__ float red[4][NTHREADS / 32];

  float sum = 0.0f, sumsq = 0.0f;
  float mn = __builtin_huge_valf(), mx = -__builtin_huge_valf();

  // Issue one 16-row stage: 912 b128 transfers; clamp index so every lane
  // issues exactly ISSUE_ITERS instructions (uniform EXEC, fixed ASYNCcnt).
  auto issue = [&](int s, int buf) {
    const float* gbase = image + (size_t)(y_low + s * ROWS) * IMG_W + xa;
#pragma unroll
    for (int k = 0; k < ISSUE_ITERS; ++k) {
      int i = tid + k * NTHREADS;
      i = (i < TPS) ? i : (TPS - 1);           // duplicate last transfer: benign
      const int r = i / (ROW_PAD / 4);         // row within stage
      const int c = i - r * (ROW_PAD / 4);     // float4 column
      async_copy_b128(gbase + (size_t)r * IMG_W + c * 4,
                      &lds[buf][r * ROW_PAD + c * 4]);
    }
  };

  issue(0, 0);
  for (int s = 0; s < STAGES; ++s) {
    if (s + 1 < STAGES) {
      issue(s + 1, (s + 1) & 1);
      WAIT_ASYNC(ISSUE_ITERS);   // only next stage in flight -> stage s landed
    } else {
      WAIT_ASYNC(0);
    }
    __syncthreads();             // all 8 waves' stage-s data visible in LDS

    const float* buf = lds[s & 1];
#pragma unroll
    for (int k = 0; k < (ROWS * PATCH) / NTHREADS; ++k) {  // 14 iters, uniform
      const int j = tid + k * NTHREADS;                    // j < 3584
      const int r = j / PATCH;
      const int c = j - r * PATCH;
      const float v = buf[r * ROW_PAD + d + c];
      sum += v;
      sumsq = fmaf(v, v, sumsq);
      mn = fminf(mn, v);
      mx = fmaxf(mx, v);
    }
    __syncthreads();             // buffer may be overwritten by stage s+2
  }

  // wave32 butterfly reduction (gfx1250 is wave32-only)
#pragma unroll
  for (int off = 16; off > 0; off >>= 1) {
    sum   += __shfl_xor(sum, off, 32);
    sumsq += __shfl_xor(sumsq, off, 32);
    mn = fminf(mn, __shfl_xor(mn, off, 32));
    mx = fmaxf(mx, __shfl_xor(mx, off, 32));
  }

  const int wave = tid >> 5;
  if ((tid & 31) == 0) {
    red[0][wave] = sum;
    red[1][wave] = sumsq;
    red[2][wave] = mn;
    red[3][wave] = mx;
  }
  __syncthreads();

  if (tid == 0) {
    float S = red[0][0], Q = red[1][0], m = red[2][0], M = red[3][0];
#pragma unroll
    for (int w = 1; w < NTHREADS / 32; ++w) {
      S += red[0][w];
      Q += red[1][w];
      m = fminf(m, red[2][w]);
      M = fmaxf(M, red[3][w]);
    }
    const float Nf   = (float)PATCH_ELEMS;
    const float mean = S / Nf;
    float var = (Q - S * mean) / (Nf - 1.0f);   // unbiased (ddof=1)
    var = var > 0.0f ? var : 0.0f;
    out[n * 4 + 0] = mean;
    out[n * 4 + 1] = sqrtf(var);
    out[n * 4 + 2] = m;
    out[n * 4 + 3] = M;
  }
}

// ---------------- launch ----------------
extern "C" void kernel_launch(void* const* d_in, const int* in_sizes, int n_in,
                              void* d_out, int out_size, void* d_ws, size_t ws_size,
                              hipStream_t stream) {
  (void)n_in; (void)out_size; (void)d_ws; (void)ws_size;
  const float* image = (const float*)d_in[0];
  const int* cx      = (const int*)d_in[1];
  const int* cy      = (const int*)d_in[2];
  float* out         = (float*)d_out;
  const int N        = in_sizes[1];   // number of nodes (2048)

  hipLaunchKernelGGL(patch_stats_kernel, dim3(N), dim3(NTHREADS), 0, stream,
                     image, cx, cy, out);
}